// TopKGroupRouter_19258633355498
// MI455X (gfx1250) — compile-verified
//
#include <hip/hip_runtime.h>
#include <hip/hip_bf16.h>
#include <stddef.h>

// ---------------------------------------------------------------------------
// TopKGroupRouter for MI455X (gfx1250, wave32).
//   groups: (G=8, B=64, C=256, H=28, W=28) f32
//   W1: (G,256,64)  b1: (G,64)  W2: (G,64)  b2: (G,)
// Outputs concatenated in d_out (f32):
//   out (64,768,28,28) | logits (64,8) | hard_mask (64,8) | soft_probs (64,8) | load_loss (1)
// Bandwidth-bound: ~720MB traffic -> ~31us at 23.3 TB/s. Gate MLP (8.4 MFLOP)
// done with V_WMMA_F32_16X16X4_F32 (exact fp32).
// ---------------------------------------------------------------------------

#define Gn 8
#define Bn 64
#define Cn 256
#define HWn 784              // 28*28
#define Hid 64
#define Kt 3
#define ROW_F4 196           // 784/4
#define CHUNK_F4 50176       // Cn*HWn/4  (one (b,k) slab in float4s)
#define BLK_PER_CHUNK 196    // CHUNK_F4 / 256

typedef __attribute__((ext_vector_type(2))) float v2f;
typedef __attribute__((ext_vector_type(4))) float v4f;   // native vector for NT load/store
typedef __attribute__((ext_vector_type(8))) float v8f;

// d_out element offsets
#define OUT_ELEMS   ((size_t)Bn * Kt * Cn * HWn)   // 38,535,168
#define LOGITS_OFF  OUT_ELEMS
#define HARD_OFF    (LOGITS_OFF + (size_t)Bn * Gn)
#define SOFT_OFF    (HARD_OFF + (size_t)Bn * Gn)
#define LOSS_OFF    (SOFT_OFF + (size_t)Bn * Gn)

// ---------------------------------------------------------------------------
// Kernel 1: spatial mean pool. One wave per (g,b,c) row of 784 contiguous f32.
// Fully coalesced 16B streaming; non-temporal (groups >> 192MB L2, re-used
// only much later by the gather which streams anyway).
// ---------------------------------------------------------------------------
__global__ __launch_bounds__(256) void pool_kernel(const float* __restrict__ groups,
                                                   float* __restrict__ pooled) {
    const int wave = threadIdx.x >> 5;
    const int lane = threadIdx.x & 31;
    const size_t ch = (size_t)blockIdx.x * 8 + wave;   // flat (g*B + b)*C + c, 131072 total
    const v4f* src = (const v4f*)(groups + ch * HWn);
    float s = 0.0f;
    #pragma unroll
    for (int j = 0; j < 7; ++j) {
        int idx = lane + j * 32;
        if (idx < ROW_F4) {
            v4f v = __builtin_nontemporal_load(&src[idx]);
            s += v.x + v.y + v.z + v.w;
        }
    }
    // wave32 butterfly reduction
    s += __shfl_xor(s, 16);
    s += __shfl_xor(s, 8);
    s += __shfl_xor(s, 4);
    s += __shfl_xor(s, 2);
    s += __shfl_xor(s, 1);
    if (lane == 0) pooled[ch] = s * (1.0f / (float)HWn);
}

// ---------------------------------------------------------------------------
// Kernel 2: per-group gate MLP via V_WMMA_F32_16X16X4_F32.
// One block per group g; 16 waves, each owns one 16x16 tile of h = relu(P@W1+b1),
// P = pooled_g (64x256), W1_g (256x64). Then 64 threads reduce h@W2 + b2 -> logits.
//
// A frag (16x4 f32, 2 VGPRs): lanes 0-15 -> M=lane, K={0,1}; lanes 16-31 -> K={2,3}.
// B frag (4x16 f32, 2 VGPRs): VGPR0 -> K=0 (lanes 0-15) / K=2 (lanes 16-31); VGPR1 -> K=1/K=3.
// C/D (16x16 f32, 8 VGPRs):  VGPRj -> row j (lanes 0-15) / row j+8 (lanes 16-31), N=lane%16.
// ---------------------------------------------------------------------------
__global__ __launch_bounds__(512) void gate_kernel(const float* __restrict__ pooled,
                                                   const float* __restrict__ W1,
                                                   const float* __restrict__ b1,
                                                   const float* __restrict__ W2,
                                                   const float* __restrict__ b2,
                                                   float* __restrict__ logits_out) {
    __shared__ float hbuf[Bn * Hid];   // 16 KB
    const int g    = blockIdx.x;
    const int lane = threadIdx.x & 31;
    const int w    = threadIdx.x >> 5;        // 0..15
    const int bm   = w >> 2;                  // row-tile 0..3
    const int hn   = w & 3;                   // col-tile 0..3

    const float* Pg  = pooled + (size_t)g * Bn * Cn;
    const float* W1g = W1     + (size_t)g * Cn * Hid;

    const int m     = lane & 15;
    const int khalf = (lane >> 4) << 1;       // 0 or 2
    const int arow  = bm * 16 + m;            // batch row this lane feeds for A
    const int bcol  = hn * 16 + m;            // hidden col this lane feeds for B

    v8f acc = {};
    for (int k0 = 0; k0 < Cn; k0 += 4) {
        v2f a, b;
        a.x = Pg[arow * Cn + k0 + khalf];
        a.y = Pg[arow * Cn + k0 + khalf + 1];
        b.x = W1g[(k0 + khalf) * Hid + bcol];
        b.y = W1g[(k0 + khalf + 1) * Hid + bcol];
        acc = __builtin_amdgcn_wmma_f32_16x16x4_f32(
            /*neg_a=*/false, a, /*neg_b=*/false, b,
            /*c_mod=*/(short)0, acc, /*reuse_a=*/false, /*reuse_b=*/false);
    }

    // bias + relu, scatter tile to LDS h[64][64]
    const int rbase = (lane >> 4) << 3;       // 0 or 8
    const int hidx  = hn * 16 + m;
    const float bias = b1[g * Hid + hidx];
    #pragma unroll
    for (int j = 0; j < 8; ++j) {
        const int bidx = bm * 16 + rbase + j;
        float v = acc[j] + bias;
        hbuf[bidx * Hid + hidx] = v > 0.0f ? v : 0.0f;
    }
    __syncthreads();

    // second projection: logits[b][g] = h[b] . W2_g + b2_g   (written as (B,G))
    if (threadIdx.x < Bn) {
        const int bb = threadIdx.x;
        float acc2 = b2[g];
        #pragma unroll 8
        for (int hh = 0; hh < Hid; ++hh)
            acc2 += hbuf[bb * Hid + hh] * W2[g * Hid + hh];
        logits_out[bb * Gn + g] = acc2;
    }
}

// ---------------------------------------------------------------------------
// Kernel 3: softmax, top-3 (ties -> lowest index, matching jax.lax.top_k),
// hard mask, selected probs, load-balance loss. One block, thread b per row.
// ---------------------------------------------------------------------------
__global__ __launch_bounds__(64) void route_kernel(const float* __restrict__ logits,
                                                   float* __restrict__ hard_mask,
                                                   float* __restrict__ soft_probs,
                                                   float* __restrict__ loss_out,
                                                   int* __restrict__ topk_idx,
                                                   float* __restrict__ sel_probs) {
    __shared__ float sp[Bn * Gn];
    const int b = threadIdx.x;

    float l[Gn], p[Gn];
    float mx = -3.402823466e38f;
    #pragma unroll
    for (int g = 0; g < Gn; ++g) { l[g] = logits[b * Gn + g]; mx = l[g] > mx ? l[g] : mx; }
    float sum = 0.0f;
    #pragma unroll
    for (int g = 0; g < Gn; ++g) { p[g] = __expf(l[g] - mx); sum += p[g]; }
    const float inv = 1.0f / sum;
    #pragma unroll
    for (int g = 0; g < Gn; ++g) {
        p[g] *= inv;
        soft_probs[b * Gn + g] = p[g];
        sp[b * Gn + g] = p[g];
    }

    float hm[Gn];
    bool used[Gn];
    #pragma unroll
    for (int g = 0; g < Gn; ++g) { hm[g] = 0.0f; used[g] = false; }
    #pragma unroll
    for (int k = 0; k < Kt; ++k) {
        float best = -3.402823466e38f; int bi = 0;
        #pragma unroll
        for (int g = 0; g < Gn; ++g)
            if (!used[g] && l[g] > best) { best = l[g]; bi = g; }
        used[bi] = true;
        hm[bi] = 1.0f;
        topk_idx[b * Kt + k]  = bi;
        sel_probs[b * Kt + k] = p[bi];
    }
    #pragma unroll
    for (int g = 0; g < Gn; ++g) hard_mask[b * Gn + g] = hm[g];

    __syncthreads();
    if (b == 0) {
        float loss = 0.0f;
        #pragma unroll
        for (int g = 0; g < Gn; ++g) {
            float s = 0.0f;
            for (int bb = 0; bb < Bn; ++bb) s += sp[bb * Gn + g];
            const float imp = s * (1.0f / (float)Bn);
            loss += imp * imp;
        }
        loss_out[0] = 0.01f * (float)Gn * loss;
    }
}

// ---------------------------------------------------------------------------
// Kernel 4: gather selected group slabs, scale by sel prob. Each (b,k) slab is
// C*784 contiguous f32 in both src and dst -> pure 16B streaming copy.
// Non-temporal stores (output never re-read on device).
// ---------------------------------------------------------------------------
__global__ __launch_bounds__(256) void gather_kernel(const float* __restrict__ groups,
                                                     const int* __restrict__ topk_idx,
                                                     const float* __restrict__ sel_probs,
                                                     float* __restrict__ out) {
    const int chunk = blockIdx.x / BLK_PER_CHUNK;                // (b*K + k)
    const int off4  = (blockIdx.x % BLK_PER_CHUNK) * 256 + threadIdx.x;
    const int g = topk_idx[chunk];
    const float s = sel_probs[chunk];
    const int b = chunk / Kt;

    const v4f* src = (const v4f*)(groups + ((size_t)(g * Bn + b)) * Cn * HWn);
    v4f v = __builtin_nontemporal_load(&src[off4]);
    v *= s;
    v4f* dst = (v4f*)(out + (size_t)chunk * Cn * HWn);
    __builtin_nontemporal_store(v, &dst[off4]);
}

// ---------------------------------------------------------------------------
extern "C" void kernel_launch(void* const* d_in, const int* in_sizes, int n_in,
                              void* d_out, int out_size, void* d_ws, size_t ws_size,
                              hipStream_t stream) {
    const float* groups = (const float*)d_in[0];
    const float* W1     = (const float*)d_in[1];
    const float* b1     = (const float*)d_in[2];
    const float* W2     = (const float*)d_in[3];
    const float* b2     = (const float*)d_in[4];

    float* out        = (float*)d_out;
    float* logits     = out + LOGITS_OFF;
    float* hard_mask  = out + HARD_OFF;
    float* soft_probs = out + SOFT_OFF;
    float* loss_out   = out + LOSS_OFF;

    // workspace: pooled (512KB) | topk_idx (768B) | sel_probs (768B)
    float* pooled  = (float*)d_ws;
    int*   topk    = (int*)((char*)d_ws + (size_t)Gn * Bn * Cn * 4);
    float* selp    = (float*)((char*)d_ws + (size_t)Gn * Bn * Cn * 4 + (size_t)Bn * Kt * 4);

    // 1) mean pool: 131072 rows, 8 waves/block
    pool_kernel<<<(Gn * Bn * Cn) / 8, 256, 0, stream>>>(groups, pooled);
    // 2) gate MLP with WMMA f32 16x16x4: one block per group
    gate_kernel<<<Gn, 512, 0, stream>>>(pooled, W1, b1, W2, b2, logits);
    // 3) softmax / top-k / mask / loss
    route_kernel<<<1, Bn, 0, stream>>>(logits, hard_mask, soft_probs, loss_out, topk, selp);
    // 4) gather + scale: B*K slabs of 196 blocks each
    gather_kernel<<<Bn * Kt * BLK_PER_CHUNK, 256, 0, stream>>>(groups, topk, selp, out);
}